// PNP_85753317032424
// MI455X (gfx1250) — compile-verified
//
#include <hip/hip_runtime.h>
#include <hip/hip_bf16.h>

typedef __attribute__((ext_vector_type(2))) float v2f;
typedef __attribute__((ext_vector_type(4))) float v4f;
typedef __attribute__((ext_vector_type(8))) float v8f;

// ---------------------------------------------------------------------------
// fp32 WMMA GEMM:  C[M,N] = A[M,K] @ B[K,N] (+ bias[N])
// Macro-tile 64x128, 8 waves (256 threads); each wave computes a 32x32 C
// block (2x2 subtiles of 16x16) so every A/B fragment feeds 2 WMMAs.
// Per K16 step each wave issues 16x v_wmma_f32_16x16x4_f32.
// Requires M % 64 == 0, N % 128 == 0, K % 16 == 0 (true for all launches).
// ---------------------------------------------------------------------------
__global__ __launch_bounds__(256) void gemm_f32_wmma(
    const float* __restrict__ A, const float* __restrict__ B,
    const float* __restrict__ bias, float* __restrict__ C,
    int M, int N, int K)
{
    constexpr int ASTR = 20;            // padded A stride: bank-spread, 16B-aligned
    __shared__ float As[64 * ASTR];     // 5 KB
    __shared__ float Bs[16 * 128];      // 8 KB

    const int tid  = threadIdx.x;
    const int lane = tid & 31;
    const int wave = tid >> 5;          // 0..7
    const int wm   = wave & 1;          // 2 waves along M (x32)
    const int wn   = wave >> 1;         // 4 waves along N (x32)
    const int hi   = lane >> 4;         // K split / row+8 split
    const int lo   = lane & 15;

    const int mBase = blockIdx.x * 64;
    const int nBase = blockIdx.y * 128;

    v8f acc00 = {}, acc01 = {}, acc10 = {}, acc11 = {};

    for (int kBase = 0; kBase < K; kBase += 16) {
        // Stage A tile 64x16 (1024 floats, one float4 per thread)
        {
            int r = tid >> 2;                 // 0..63
            int c = (tid & 3) * 4;            // 0,4,8,12
            v4f av = *(const v4f*)&A[(size_t)(mBase + r) * K + kBase + c];
            *(v4f*)&As[r * ASTR + c] = av;
        }
        // Stage B tile 16x128 (2048 floats, two float4 per thread)
        {
            int r = tid >> 5;                 // 0..7
            int c = (tid & 31) * 4;           // 0..124
            v4f b0 = *(const v4f*)&B[(size_t)(kBase + r)     * N + nBase + c];
            v4f b1 = *(const v4f*)&B[(size_t)(kBase + r + 8) * N + nBase + c];
            *(v4f*)&Bs[r * 128 + c]       = b0;
            *(v4f*)&Bs[(r + 8) * 128 + c] = b1;
        }
        __syncthreads();

#pragma unroll
        for (int kk = 0; kk < 16; kk += 4) {
            // Two A fragments (rows wm*32 + {0,16} + lo), K chunk kk
            v2f a0, a1, b0, b1;
            a0.x = As[(wm * 32 +      lo) * ASTR + kk + 2 * hi];
            a0.y = As[(wm * 32 +      lo) * ASTR + kk + 2 * hi + 1];
            a1.x = As[(wm * 32 + 16 + lo) * ASTR + kk + 2 * hi];
            a1.y = As[(wm * 32 + 16 + lo) * ASTR + kk + 2 * hi + 1];
            // Two B fragments (cols wn*32 + {0,16} + lo)
            b0.x = Bs[(kk + 2 * hi)     * 128 + wn * 32 +      lo];
            b0.y = Bs[(kk + 2 * hi + 1) * 128 + wn * 32 +      lo];
            b1.x = Bs[(kk + 2 * hi)     * 128 + wn * 32 + 16 + lo];
            b1.y = Bs[(kk + 2 * hi + 1) * 128 + wn * 32 + 16 + lo];

            acc00 = __builtin_amdgcn_wmma_f32_16x16x4_f32(false, a0, false, b0, (short)0, acc00, false, false);
            acc01 = __builtin_amdgcn_wmma_f32_16x16x4_f32(false, a0, false, b1, (short)0, acc01, false, false);
            acc10 = __builtin_amdgcn_wmma_f32_16x16x4_f32(false, a1, false, b0, (short)0, acc10, false, false);
            acc11 = __builtin_amdgcn_wmma_f32_16x16x4_f32(false, a1, false, b1, (short)0, acc11, false, false);
        }
        __syncthreads();
    }

    // C/D layout: vgpr r -> rows r (lanes 0-15) / r+8 (lanes 16-31), col = lo
    const int col0 = nBase + wn * 32 + lo;
    const int col1 = col0 + 16;
    const float bv0 = bias ? bias[col0] : 0.0f;
    const float bv1 = bias ? bias[col1] : 0.0f;
#pragma unroll
    for (int r = 0; r < 8; ++r) {
        int row0 = mBase + wm * 32 + r + 8 * hi;
        int row1 = row0 + 16;
        C[(size_t)row0 * N + col0] = acc00[r] + bv0;
        C[(size_t)row0 * N + col1] = acc01[r] + bv1;
        C[(size_t)row1 * N + col0] = acc10[r] + bv0;
        C[(size_t)row1 * N + col1] = acc11[r] + bv1;
    }
}

// ---------------------------------------------------------------------------
// PnP flash attention + alpha residual blend.
//   q,k: [SB=2, S, D] (source batches only; batch b reads b % 2)
//   v,x: [B=6, S, D]
//   o  : [B, S, D] = 0.5 * softmax(q k^T / sqrt(dh)) v  +  0.5 * x
// One wave per (b, h, 16-row q tile); online softmax over 64 key tiles.
// ---------------------------------------------------------------------------
__global__ __launch_bounds__(32) void pnp_attn_f32_wmma(
    const float* __restrict__ q, const float* __restrict__ k,
    const float* __restrict__ v, const float* __restrict__ x,
    float* __restrict__ o)
{
    constexpr int S = 1024, D = 640, DH = 80, SB = 2;
    const float scale = 0.1118033988749895f;  // 80^-0.5

    __shared__ float Plds[16 * 16];           // P tile C-layout -> A-layout bounce

    const int lane = threadIdx.x;
    const int hi = lane >> 4, lo = lane & 15;
    const int qt = blockIdx.x;                // 0..63 query tile
    const int h  = blockIdx.y;                // 0..7
    const int b  = blockIdx.z;                // 0..5
    const int bsrc = b % SB;                  // PnP: tile(q[:2], 3)

    const float* qb = q + (size_t)bsrc * S * D + h * DH;
    const float* kb = k + (size_t)bsrc * S * D + h * DH;
    const float* vb = v + (size_t)b    * S * D + h * DH;

    // Q fragments: 16 rows x 80 cols as 20 K4-chunks, scale folded in.
    v2f qa[20];
    {
        const float* qr = qb + (size_t)(qt * 16 + lo) * D;
#pragma unroll
        for (int c = 0; c < 20; ++c) {
            v2f t = *(const v2f*)(qr + c * 4 + 2 * hi);
            qa[c].x = t.x * scale;
            qa[c].y = t.y * scale;
        }
    }

    v8f acc[5] = {};                          // out tile 16 x 80 (5 col tiles)
    float m_run[8], l_run[8];
#pragma unroll
    for (int r = 0; r < 8; ++r) { m_run[r] = -1e30f; l_run[r] = 0.0f; }

    for (int kt = 0; kt < S / 16; ++kt) {
        // ---- S = (q*scale) @ k^T : K^T B-frag loads like an A-frag of k ----
        v8f s = {};
        const float* kr = kb + (size_t)(kt * 16 + lo) * D;
#pragma unroll
        for (int c = 0; c < 20; ++c) {
            v2f bf = *(const v2f*)(kr + c * 4 + 2 * hi);
            s = __builtin_amdgcn_wmma_f32_16x16x4_f32(
                false, qa[c], false, bf, (short)0, s, false, false);
        }

        // ---- online softmax: row stats live replicated per 16-lane half ----
#pragma unroll
        for (int r = 0; r < 8; ++r) {
            float tm = s[r];
            tm = fmaxf(tm, __shfl_xor(tm, 1, 32));
            tm = fmaxf(tm, __shfl_xor(tm, 2, 32));
            tm = fmaxf(tm, __shfl_xor(tm, 4, 32));
            tm = fmaxf(tm, __shfl_xor(tm, 8, 32));
            float mn   = fmaxf(m_run[r], tm);
            float corr = __expf(m_run[r] - mn);
            float p    = __expf(s[r] - mn);
            float ts = p;
            ts += __shfl_xor(ts, 1, 32);
            ts += __shfl_xor(ts, 2, 32);
            ts += __shfl_xor(ts, 4, 32);
            ts += __shfl_xor(ts, 8, 32);
            l_run[r] = l_run[r] * corr + ts;
            m_run[r] = mn;
            s[r] = p;
#pragma unroll
            for (int t = 0; t < 5; ++t) acc[t][r] *= corr;
        }

        // ---- convert P from C-layout to A-layout via LDS ----
#pragma unroll
        for (int r = 0; r < 8; ++r) Plds[(r + 8 * hi) * 16 + lo] = s[r];
        __syncthreads();
        v2f pa[4];
#pragma unroll
        for (int kk = 0; kk < 4; ++kk) {
            pa[kk].x = Plds[lo * 16 + kk * 4 + 2 * hi];
            pa[kk].y = Plds[lo * 16 + kk * 4 + 2 * hi + 1];
        }
        __syncthreads();

        // ---- acc += P @ V  (K=16 keys as 4 chunks, 5 dh column tiles) ----
#pragma unroll
        for (int kk = 0; kk < 4; ++kk) {
            const float* vr = vb + (size_t)(kt * 16 + kk * 4 + 2 * hi) * D;
#pragma unroll
            for (int t = 0; t < 5; ++t) {
                v2f bf;
                bf.x = vr[t * 16 + lo];
                bf.y = vr[D + t * 16 + lo];
                acc[t] = __builtin_amdgcn_wmma_f32_16x16x4_f32(
                    false, pa[kk], false, bf, (short)0, acc[t], false, false);
            }
        }
    }

    // ---- normalize, alpha-blend residual (ALPHA = 0.5), store ----
#pragma unroll
    for (int r = 0; r < 8; ++r) {
        int row = qt * 16 + r + 8 * hi;
        float inv = 1.0f / l_run[r];
        const float* xr = x + ((size_t)b * S + row) * D + h * DH;
        float* orow     = o + ((size_t)b * S + row) * D + h * DH;
#pragma unroll
        for (int t = 0; t < 5; ++t) {
            float val = acc[t][r] * inv;
            orow[t * 16 + lo] = 0.5f * val + 0.5f * xr[t * 16 + lo];
        }
    }
}

// ---------------------------------------------------------------------------
// Launch: q/k projections for source batches only (M=2048), v for all (6144),
// flash attention + blend, then output projection with bias.
// ---------------------------------------------------------------------------
extern "C" void kernel_launch(void* const* d_in, const int* in_sizes, int n_in,
                              void* d_out, int out_size, void* d_ws, size_t ws_size,
                              hipStream_t stream)
{
    (void)in_sizes; (void)n_in; (void)out_size; (void)ws_size;

    const float* x  = (const float*)d_in[0];   // [6,1024,640]
    const float* Wq = (const float*)d_in[1];   // [640,640]
    const float* Wk = (const float*)d_in[2];
    const float* Wv = (const float*)d_in[3];
    const float* Wo = (const float*)d_in[4];
    const float* bo = (const float*)d_in[5];   // [640]
    float* out = (float*)d_out;                // [6,1024,640]

    constexpr int S = 1024, D = 640, B = 6, SB = 2;
    constexpr size_t QK_ELEMS = (size_t)SB * S * D;  // 1,310,720
    constexpr size_t V_ELEMS  = (size_t)B  * S * D;  // 3,932,160

    float* ws  = (float*)d_ws;
    float* wq_ = ws;                 // q  [2,1024,640]
    float* wk_ = wq_ + QK_ELEMS;     // k  [2,1024,640]
    float* wv_ = wk_ + QK_ELEMS;     // v  [6,1024,640]
    float* wo_ = wv_ + V_ELEMS;      // blended attn out [6,1024,640]

    dim3 gblk(256);
    // q, k for source batches only (rows 0..2047 of x are batches 0,1)
    gemm_f32_wmma<<<dim3((SB * S) / 64, D / 128), gblk, 0, stream>>>(
        x, Wq, nullptr, wq_, SB * S, D, D);
    gemm_f32_wmma<<<dim3((SB * S) / 64, D / 128), gblk, 0, stream>>>(
        x, Wk, nullptr, wk_, SB * S, D, D);
    // v for all batches
    gemm_f32_wmma<<<dim3((B * S) / 64, D / 128), gblk, 0, stream>>>(
        x, Wv, nullptr, wv_, B * S, D, D);

    // flash attention + residual blend
    pnp_attn_f32_wmma<<<dim3(S / 16, 8, B), dim3(32), 0, stream>>>(
        wq_, wk_, wv_, x, wo_);

    // output projection with bias
    gemm_f32_wmma<<<dim3((B * S) / 64, D / 128), gblk, 0, stream>>>(
        wo_, Wo, bo, out, B * S, D, D);
}